// Model_63239098466794
// MI455X (gfx1250) — compile-verified
//
#include <hip/hip_runtime.h>
#include <stdint.h>
#include <stddef.h>

typedef uint32_t u32x4 __attribute__((ext_vector_type(4)));
typedef uint32_t u32x8 __attribute__((ext_vector_type(8)));

namespace {
constexpr int kNZ     = 131072;
constexpr int kNT     = 2048;
constexpr int kNOut   = 16;                  // steps per snapshot chunk
constexpr int kTile   = 1024;                // points owned per block
constexpr int kCPS    = 8;                   // chunks per superstep (128 steps)
constexpr int kHalo   = kNOut * kCPS;        // 128
constexpr int kR      = kTile + 2 * kHalo;   // 1280 LDS row span
constexpr int kNB     = kNZ / kTile;         // 128 blocks
constexpr int kNTh    = 256;                 // 8 waves (wave32)
constexpr int kEPT    = kR / kNTh;           // 5 elements / thread
constexpr int kNChunk = kNT / kNOut;         // 128 snapshots
constexpr int kNSS    = kNChunk / kCPS;      // 16 launches
constexpr int kSH     = 4;                   // 16B zero guard pad each side
}

__device__ __forceinline__ uint32_t lds_off(const void* p) {
  // flat addr of __shared__ has LDS offset in its low 32 bits
  return (uint32_t)(uintptr_t)p;
}

__global__ __launch_bounds__(kNTh) void diffusion_superstep(
    const float* __restrict__ u_in,   // current state [NZ]
    const float* __restrict__ hz,     // grid spacing  [NZ]
    float* __restrict__ u_next,       // state after 128 steps [NZ]
    float* __restrict__ snap_out,     // d_out snapshot base [128][NZ]
    int chunk_base)
{
  __shared__ float ub[2][kR + 2 * kSH];   // padded ping-pong rows (16B-aligned stride)
  __shared__ float cc[kR];                // per-point coefficient (0 outside domain)

  const int tid = (int)threadIdx.x;
  const int t0  = (int)blockIdx.x * kTile;
  const int g0  = t0 - kHalo;             // global index of LDS slot 0
  const float coef = (float)(0.5 * 0.4 * 0.1 * 1.0);   // alpha*beta*dt*flx

  // ---- 1. TDM: DMA the in-domain span of (tile+halo) into LDS as a 1-row tensor
  {
    const int sg = (g0 < 0) ? 0 : g0;                       // clamped span start
    const int eg = (g0 + kR > kNZ) ? kNZ : (g0 + kR);       // clamped span end
    const uint32_t cnt  = (uint32_t)(eg - sg);              // elements to DMA
    const uint32_t lds0 = lds_off(&ub[0][kSH + (sg - g0)]);
    const uint64_t ga   = (uint64_t)(uintptr_t)(u_in + sg); // byte address
    const u32x4 d0 = { 1u,                                   // count=1 valid D#
                       lds0,                                 // lds_addr
                       (uint32_t)ga,                         // global_addr[31:0]
                       ((uint32_t)(ga >> 32) & 0x1FFFFFFu) | (2u << 30) }; // [56:32]|type=2
    const u32x8 d1 = { 2u << 16,                 // data_size=4B; mask/pad/iterate=0
                       (cnt & 0xFFFFu) << 16,    // tensor_dim0[15:0]
                       (cnt >> 16) | (1u << 16), // tensor_dim0[31:16] | tensor_dim1=1
                       cnt << 16,                // tensor_dim1[31:16]=0 | tile_dim0=cnt
                       1u,                       // tile_dim1=1, tile_dim2=0
                       cnt,                      // tensor_dim0_stride (single row)
                       0u, 0u };
    if (tid == 0) {  // one DMA per block (TDM ignores EXEC; issue from wave 0 only)
      asm volatile("tensor_load_to_lds %0, %1" :: "s"(d0), "s"(d1) : "memory");
    }
  }

  // ---- 2. async global->LDS: raw hz row into cc[]; zero out-of-domain u slots
  for (int j = 0; j < kEPT; ++j) {
    const int l = tid + j * kNTh;
    const int g = g0 + l;
    if (g >= 0 && g < kNZ) {
      asm volatile("global_load_async_to_lds_b32 %0, %1, %2"
                   :: "v"(lds_off(&cc[l])), "v"((uint32_t)g * 4u), "s"(hz)
                   : "memory");
    } else {
      ub[0][kSH + l] = 0.0f;   // zero-pad region: stays 0 forever (cc=0 below)
    }
  }
  if (tid < kSH) {             // zero the guard pads of both buffers
    ub[0][tid] = 0.0f;  ub[0][kSH + kR + tid] = 0.0f;
    ub[1][tid] = 0.0f;  ub[1][kSH + kR + tid] = 0.0f;
  }
  asm volatile("s_wait_asynccnt 0" ::: "memory");   // this wave's hz loads landed
  for (int j = 0; j < kEPT; ++j) {                  // in-place: hz -> coefficient
    const int l = tid + j * kNTh;
    const int g = g0 + l;
    if (g >= 0 && g < kNZ) { const float h = cc[l]; cc[l] = coef / (h * h); }
    else                   { cc[l] = 0.0f; }
  }
  if (tid == 0) __builtin_amdgcn_s_wait_tensorcnt(0);  // TDM tile landed (wave 0)
  __syncthreads();

  // ---- 3. 8 chunks x 16 steps, branch-free, entirely in LDS ----
  int p = 0;
  for (int k = 0; k < kCPS; ++k) {
    {  // snapshot of state BEFORE advancing this chunk (coalesced b128)
      float* dst  = snap_out + (size_t)(chunk_base + k) * (size_t)kNZ + (size_t)t0;
      const int j = tid * 4;                     // 256 threads * 4 = kTile
      const float4 v = *reinterpret_cast<const float4*>(&ub[p][kSH + kHalo + j]);
      *reinterpret_cast<float4*>(dst + j) = v;
    }
    for (int s = 0; s < kNOut; ++s) {
      const int q = p ^ 1;
#pragma unroll
      for (int j = 0; j < kEPT; ++j) {
        const int l  = tid + j * kNTh;
        const float um = ub[p][kSH + l - 1];
        const float uc = ub[p][kSH + l];
        const float up = ub[p][kSH + l + 1];
        ub[q][kSH + l] = uc + cc[l] * (up - 2.0f * uc + um);
      }
      __syncthreads();
      p = q;
    }
  }

  // ---- 4. publish final state via async LDS->global (b128, 16B aligned) ----
  {
    const uint32_t lds0 = lds_off(&ub[p][kSH + kHalo + tid * 4]);
    const uint32_t goff = (uint32_t)(t0 + tid * 4) * 4u;
    asm volatile("global_store_async_from_lds_b128 %0, %1, %2"
                 :: "v"(goff), "v"(lds0), "s"(u_next)
                 : "memory");
    asm volatile("s_wait_asynccnt 0" ::: "memory");
  }
}

__global__ void write_time_kernel(float* __restrict__ t) {
  const int k = (int)threadIdx.x;
  if (k < kNChunk) t[k] = (float)((double)k * 1.6);
}

extern "C" void kernel_launch(void* const* d_in, const int* in_sizes, int n_in,
                              void* d_out, int out_size, void* d_ws, size_t ws_size,
                              hipStream_t stream) {
  (void)in_sizes; (void)n_in; (void)out_size; (void)ws_size;

  const float* u0 = (const float*)d_in[0];
  const float* hz = (const float*)d_in[1];
  float* out = (float*)d_out;
  float* uA  = (float*)d_ws;        // ping-pong state buffers (2 * 512 KB)
  float* uB  = uA + kNZ;

  // time vector lives after the 128 snapshots
  write_time_kernel<<<1, kNChunk, 0, stream>>>(out + (size_t)kNChunk * (size_t)kNZ);

  for (int ss = 0; ss < kNSS; ++ss) {
    const float* in = (ss == 0) ? u0 : ((ss & 1) ? uA : uB);
    float* nxt      = (ss & 1) ? uB : uA;
    diffusion_superstep<<<kNB, kNTh, 0, stream>>>(in, hz, nxt, out, ss * kCPS);
  }
}